// LTIRouter_60430189854723
// MI455X (gfx1250) — compile-verified
//
#include <hip/hip_runtime.h>
#include <hip/hip_bf16.h>

// Problem constants (from reference): B=1, N=1024, T=2048, E=16384, K=32
#define PN 1024
#define PT 2048
#define PE 16384
#define PK 32

typedef __attribute__((ext_vector_type(2))) float v2f;
typedef __attribute__((ext_vector_type(8))) float v8f;

// ---------------- out = x (identity term) ----------------
__global__ void lti_init_out(const float4* __restrict__ x4, float4* __restrict__ o4, int n4) {
    int i = blockIdx.x * blockDim.x + threadIdx.x;
    if (i < n4) o4[i] = x4[i];
}

// ---------------- binning of edges by src ----------------
__global__ void lti_zero(int* __restrict__ p, int n) {
    int i = blockIdx.x * blockDim.x + threadIdx.x;
    if (i < n) p[i] = 0;
}

__global__ void lti_hist(const int* __restrict__ src, int* __restrict__ counts) {
    int e = blockIdx.x * blockDim.x + threadIdx.x;
    if (e < PE) atomicAdd(&counts[src[e]], 1);
}

__global__ void lti_scan(const int* __restrict__ counts, int* __restrict__ offsets) {
    if (blockIdx.x == 0 && threadIdx.x == 0) {
        int acc = 0;
        for (int i = 0; i < PN; ++i) { offsets[i] = acc; acc += counts[i]; }
    }
}

__global__ void lti_scatter(const int* __restrict__ src, const int* __restrict__ offsets,
                            int* __restrict__ cursor, int* __restrict__ bins) {
    int e = blockIdx.x * blockDim.x + threadIdx.x;
    if (e < PE) {
        int s = src[e];
        int p = atomicAdd(&cursor[s], 1);
        bins[offsets[s] + p] = e;
    }
}

// ---------------- main FIR + scatter-add kernel ----------------
// grid = (N src rows, T/64 time tiles), block = 128 threads (4 waves).
// Each wave owns a 16-timestep sub-tile and loops over groups of 16 edges
// sharing this src row. Fully branch-free, software-pipelined inner loop:
//   issue 8x global_load_b64 of kernel taps  (padded cols read a zero pad)
//   prefetch next group's edge id + dst row  (overlaps the WMMA chain)
//   8x V_WMMA_F32_16X16X4_F32 fp32 accumulation chain over the 32 taps
//   8x global_atomic_add_f32 scatter (padded cols add exactly 0.0 -> no-op)
__global__ __launch_bounds__(128) void lti_main(
    const float* __restrict__ x, const float* __restrict__ kern,
    const int* __restrict__ dst, const int* __restrict__ bins,
    const int* __restrict__ counts, const int* __restrict__ offsets,
    const float* __restrict__ zpad, float* __restrict__ out)
{
    __shared__ float xs[96];   // x[s, t0-31 .. t0+63]

    const int s   = blockIdx.x;
    const int t0  = blockIdx.y * 64;
    const int tid = threadIdx.x;

    if (tid < 95) {
        int t = t0 - 31 + tid;
        xs[tid] = (t >= 0) ? x[s * PT + t] : 0.0f;
    }
    __syncthreads();

    const int cnt = counts[s];
    const int off = offsets[s];

    const int lane = tid & 31;
    const int woff = (tid >> 5) << 4;     // wave's time offset: 0,16,32,48
    const int m    = lane & 15;           // A row / D column index
    const int half = lane >> 4;           // 0: K 0..1 / M 0..7, 1: K 2..3 / M 8..15
    const int kb   = half << 1;
    const int abase = 31 + woff + m - kb; // LDS index of A[m, kb] at j0=0
    const int tout  = t0 + woff + half * 8;

    // A pairs are loop-invariant for this wave: hoist all 8 into registers.
    v2f av[8];
    #pragma unroll
    for (int st = 0; st < 8; ++st) {
        av[st][0] = xs[abase - 4 * st];       // A[m, kb]   at j0=4*st
        av[st][1] = xs[abase - 4 * st - 1];   // A[m, kb+1]
    }

    if (cnt > 0) {                            // uniform per block: no divergence
        // Prologue: edge id / dst row for group 0 (clamped -> always in range).
        bool valid = m < cnt;
        int  ei    = bins[off + (valid ? m : cnt - 1)];
        int  d     = dst[ei];

        for (int g0 = 0; g0 < cnt; g0 += 16) {
            // Invalid columns read zeros -> their D column is exactly 0.
            const float* wrow = valid ? (kern + ei * PK + kb) : (zpad + kb);
            v2f bv[8];
            #pragma unroll
            for (int st = 0; st < 8; ++st)
                bv[st] = *(const v2f*)(wrow + 4 * st); // B[kb+v,n'] = kernel[e,4st+kb+v]

            // Prefetch next group's edge id + dst row; latency hides under WMMAs.
            const int  ncol   = g0 + 16 + m;
            const bool nvalid = ncol < cnt;
            const int  nei    = bins[off + (nvalid ? ncol : cnt - 1)];
            const int  nd     = dst[nei];

            v8f acc = {0.f, 0.f, 0.f, 0.f, 0.f, 0.f, 0.f, 0.f};
            #pragma unroll
            for (int st = 0; st < 8; ++st)
                acc = __builtin_amdgcn_wmma_f32_16x16x4_f32(
                    false, av[st], false, bv[st], (short)0, acc, false, false);

            // D layout: lane holds column n'=m, rows M = g + 8*half (g = vgpr idx).
            // Unconditional: padded columns contribute exactly +0.0f.
            float* orow = out + d * PT + tout;
            #pragma unroll
            for (int g = 0; g < 8; ++g)
                unsafeAtomicAdd(orow + g, acc[g]);

            ei = nei; d = nd; valid = nvalid;
        }
    }
}

extern "C" void kernel_launch(void* const* d_in, const int* in_sizes, int n_in,
                              void* d_out, int out_size, void* d_ws, size_t ws_size,
                              hipStream_t stream) {
    const float* x    = (const float*)d_in[0];   // (1,1024,2048) f32
    const float* kern = (const float*)d_in[1];   // (16384,32) f32
    const int*   src  = (const int*)d_in[2];     // (16384,)
    const int*   dst  = (const int*)d_in[3];     // (16384,)
    float*       out  = (float*)d_out;           // (1,1024,2048) f32

    // workspace layout (ints):
    //   counts[1024] @0 | cursor[1024] @1024 | zpad[32] @2048 |
    //   offsets[1024] @2080 | bins[16384] @3104
    int* wsi     = (int*)d_ws;
    int* counts  = wsi;
    int* cursor  = wsi + PN;
    float* zpad  = (float*)(wsi + 2 * PN);
    int* offsets = wsi + 2 * PN + PK;
    int* bins    = wsi + 3 * PN + PK;

    // 1) out = x
    int n4 = (PN * PT) / 4;
    lti_init_out<<<(n4 + 255) / 256, 256, 0, stream>>>((const float4*)x, (float4*)out, n4);

    // 2) bin edges by src (zero pass covers counts + cursor + zpad contiguously)
    lti_zero<<<(2 * PN + PK + 255) / 256, 256, 0, stream>>>(counts, 2 * PN + PK);
    lti_hist<<<PE / 256, 256, 0, stream>>>(src, counts);
    lti_scan<<<1, 1, 0, stream>>>(counts, offsets);
    lti_scatter<<<PE / 256, 256, 0, stream>>>(src, offsets, cursor, bins);

    // 3) FIR via fp32 WMMA + atomic scatter into dst rows
    dim3 grid(PN, PT / 64);
    lti_main<<<grid, 128, 0, stream>>>(x, kern, dst, bins, counts, offsets, zpad, out);
}